// ModuleRenderScatter_12601434046904
// MI455X (gfx1250) — compile-verified
//
#include <hip/hip_runtime.h>
#include <stdint.h>
#include <limits.h>

// Problem constants (from reference setup_inputs)
#define B_N 4
#define C_N 3
#define H_N 512
#define W_N 1024
#define P_R 7                     // max splat radius
#define TILE 32                   // 32x32 output tile per block
#define HALO (TILE + 2 * P_R)     // 46
#define PITCH 48                  // padded LDS row pitch (pixels)
#define NTH 256                   // 8 wave32 per block
#define HWN (H_N * W_N)

// --- CDNA5 async global->LDS staging (guarded; sync fallback otherwise) ---
#if defined(__has_builtin)
#  if __has_builtin(__builtin_amdgcn_global_load_async_to_lds_b32) && \
      __has_builtin(__builtin_amdgcn_s_wait_asynccnt)
#    define USE_ASYNC 1
#  endif
#endif
#ifndef USE_ASYNC
#  define USE_ASYNC 0
#endif

// Builtin signature (from hipcc diagnostic): params are int* with AS1 (global,
// printed as "__device__ int *") and AS3 (LDS) pointees.
typedef __attribute__((address_space(1))) int gint_t;   // global int
typedef __attribute__((address_space(3))) int lint_t;   // LDS int

__global__ __launch_bounds__(NTH) void bokeh_gather_kernel(
    const float* __restrict__ image,
    const float* __restrict__ defocus,
    float* __restrict__ out_bokeh,
    float* __restrict__ out_dilate)
{
  // Interleaved halo tile: {img_c0, img_c1, img_c2, defocus} per pixel.
  // defocus == NaN marks out-of-image pixels: every compare below is written
  // NaN-false, exactly matching the reference's zero-padded intr / INT_MIN
  // padded defint semantics (padding never contributes).
  __shared__ float4 tile[HALO * PITCH];   // 46*48*16 = 35,328 B

  const int tid = threadIdx.x;
  const int b   = blockIdx.z;
  const int ty0 = blockIdx.y * TILE;
  const int tx0 = blockIdx.x * TILE;

  const float* img_b = image   + (size_t)b * C_N * HWN;
  const float* def_b = defocus + (size_t)b * HWN;

  const float QNAN = __builtin_nanf("");

  // ---- Stage halo into LDS ----
  for (int idx = tid; idx < HALO * HALO; idx += NTH) {
    const int row  = idx / HALO;
    const int col  = idx - row * HALO;
    const int gy   = ty0 - P_R + row;
    const int gx   = tx0 - P_R + col;
    const int slot = row * PITCH + col;
    if ((unsigned)gy < (unsigned)H_N && (unsigned)gx < (unsigned)W_N) {
      const size_t g = (size_t)gy * W_N + gx;
#if USE_ASYNC
      // generic->AS3: LLVM lowers this addrspacecast as truncation to the
      // low 32 bits, so the integer detour reproduces the official lowering.
      const uint32_t l0 = (uint32_t)(uintptr_t)(float*)&tile[slot];
      __builtin_amdgcn_global_load_async_to_lds_b32(
          (gint_t*)(uintptr_t)(img_b + g),           (lint_t*)l0,          0, 0);
      __builtin_amdgcn_global_load_async_to_lds_b32(
          (gint_t*)(uintptr_t)(img_b + g + HWN),     (lint_t*)(l0 + 4u),   0, 0);
      __builtin_amdgcn_global_load_async_to_lds_b32(
          (gint_t*)(uintptr_t)(img_b + g + 2 * HWN), (lint_t*)(l0 + 8u),   0, 0);
      __builtin_amdgcn_global_load_async_to_lds_b32(
          (gint_t*)(uintptr_t)(def_b + g),           (lint_t*)(l0 + 12u),  0, 0);
#else
      float4 v;
      v.x = img_b[g];
      v.y = img_b[g + HWN];
      v.z = img_b[g + 2 * HWN];
      v.w = def_b[g];
      tile[slot] = v;
#endif
    } else {
      tile[slot] = make_float4(0.f, 0.f, 0.f, QNAN);
    }
  }
#if USE_ASYNC
  __builtin_amdgcn_s_wait_asynccnt(0);
#endif
  __syncthreads();

  // ---- Gather: each thread owns a 4-row vertical strip of outputs ----
  const int lx  = tid & (TILE - 1);   // 0..31 (== lane id -> coalesced LDS/global)
  const int ly0 = (tid >> 5) * 4;     // strip base row in tile

  float wc[4]     = {0.f, 0.f, 0.f, 0.f};
  float acc[4][3] = {};
  int   dd[4]     = {INT_MIN, INT_MIN, INT_MIN, INT_MIN};

  const float K = 11.541560327111707f;  // 8 * log2(e):  0.5+0.5*tanh(4z) = 1 - 1/(1+exp2(K*z))

  for (int t = 0; t < 18; ++t) {        // source rows ly0-7 .. ly0+10
    // dy for output i at this source row: dy = i + 7 - t  (constant per t,i)
    float adym1[4], dy2[4];
#pragma unroll
    for (int i = 0; i < 4; ++i) {
      const float d = (float)(i + 7 - t);
      adym1[i] = fabsf(d) - 1.0f;
      dy2[i]   = d * d;
    }
    const float adymin1 =
        fminf(fminf(adym1[0], adym1[1]), fminf(adym1[2], adym1[3]));

    const float4* rowp = &tile[(ly0 + t) * PITCH + lx];

#pragma unroll
    for (int j = 0; j < 15; ++j) {      // source cols lx-7 .. lx+7 ; dx = 7 - j
      const float4 v = rowp[j];         // one ds_load_b128 serves 4 outputs
      const float r = fabsf(v.w);       // NaN for out-of-image sources

      const float adx   = (float)((j < 7) ? (7 - j) : (j - 7));
      const float adxm1 = adx - 1.0f;
      const float dx2   = adx * adx;

      // Wave-uniform cull: skip if no lane can have any in-footprint output.
      // thr = min_i (max(|dx|,|dy_i|) - 1); foot requires r >= thr.
      const float thr = fmaxf(adxm1, adymin1);
      if (__builtin_amdgcn_ballot_w32(r >= thr) == 0u) continue;

      const float inv_rr = __builtin_amdgcn_rcpf(fmaxf(r * r, 1.0f));
      const float Kr     = K * r;
      const int   di     = (int)v.w;    // trunc toward zero, matches astype(int32)

#pragma unroll
      for (int i = 0; i < 4; ++i) {
        const float dist = __builtin_amdgcn_sqrtf(dx2 + dy2[i]);
        // in_foot <=> floor(r)+1 >= max(|dx|,|dy|) <=> r >= m-1 (exact, NaN->false)
        const bool foot = r >= fmaxf(adxm1, adym1[i]);
        const float e = __builtin_amdgcn_exp2f(fmaf(-K, dist, Kr));
        const float u = __builtin_amdgcn_rcpf(e + 1.0f);
        float w = (1.0f - u) * inv_rr;  // (0.5+0.5*tanh(4(r-dist)))/max(r^2,1)
        w = foot ? w : 0.0f;
        wc[i]     += w;
        acc[i][0]  = fmaf(w, v.x, acc[i][0]);
        acc[i][1]  = fmaf(w, v.y, acc[i][1]);
        acc[i][2]  = fmaf(w, v.z, acc[i][2]);
        // dist <= r implies in_foot (Chebyshev <= Euclid); NaN-safe false.
        const bool circ = dist <= r;
        dd[i] = max(dd[i], circ ? di : INT_MIN);
      }
    }
  }

  // ---- Write outputs (offset (0,0) always in-circle, so dd covers self) ----
  const int x = tx0 + lx;
#pragma unroll
  for (int i = 0; i < 4; ++i) {
    const int   y = ty0 + ly0 + i;
    const size_t p = (size_t)y * W_N + x;
    const float inv = 1.0f / wc[i];     // wc > 0 always (self term)
    out_bokeh[((size_t)b * C_N + 0) * HWN + p] = acc[i][0] * inv;
    out_bokeh[((size_t)b * C_N + 1) * HWN + p] = acc[i][1] * inv;
    out_bokeh[((size_t)b * C_N + 2) * HWN + p] = acc[i][2] * inv;
    out_dilate[(size_t)b * HWN + p] = (float)dd[i];
  }
}

extern "C" void kernel_launch(void* const* d_in, const int* in_sizes, int n_in,
                              void* d_out, int out_size, void* d_ws, size_t ws_size,
                              hipStream_t stream) {
  (void)in_sizes; (void)n_in; (void)out_size; (void)d_ws; (void)ws_size;
  const float* image   = (const float*)d_in[0];
  const float* defocus = (const float*)d_in[1];
  float* out_bokeh  = (float*)d_out;
  float* out_dilate = out_bokeh + (size_t)B_N * C_N * HWN;

  dim3 grid(W_N / TILE, H_N / TILE, B_N);   // 32 x 16 x 4 = 2048 blocks
  hipLaunchKernelGGL(bokeh_gather_kernel, grid, dim3(NTH), 0, stream,
                     image, defocus, out_bokeh, out_dilate);
}